// SPWireframeDescriptor_3917010174424
// MI455X (gfx1250) — compile-verified
//
#include <hip/hip_runtime.h>
#include <hip/hip_bf16.h>

typedef float v2f __attribute__((ext_vector_type(2)));
typedef float v8f __attribute__((ext_vector_type(8)));

// ---------------- problem constants ----------------
#define N_LINES   2048
#define NEP       4096          // endpoints = N_LINES*2
#define NKP       2048          // keypoints
#define NCH       256           // descriptor channels
#define HC        128
#define WC        128
#define NALL      (NEP + NKP)   // 6144 output points
#define ADJ_WORDS 128           // 4096 bits / 32

// d_out layout (floats), reference tuple order
#define OUT_POINTS    0                       // (6144,2)   12288
#define OUT_SCORES    12288                   // 6144
#define OUT_DESCS     18432                   // 256*6144   1572864
#define OUT_NEWLINES  1591296                 // 8192
#define OUT_IDX       1599488                 // 4096
#define OUT_LS        1603584                 // 2048

// workspace layout (bytes)
#define WS_ADJ    0                           // 4096*128 u32 = 2 MB
#define WS_SUP    (WS_ADJ + NEP*ADJ_WORDS*4)  // 2048 u32
#define WS_LAB0   (WS_SUP + NKP*4)
#define WS_LAB1   (WS_LAB0 + NEP*4)
#define WS_RANK   (WS_LAB1 + NEP*4)
#define WS_CLUS   (WS_RANK + NEP*4)
#define WS_JUNC   (WS_CLUS + NEP*4)           // 4096*2 f32

// ---------------- utility kernels ----------------
__global__ void zero_kernel(unsigned* p, int n) {
    for (int i = blockIdx.x * blockDim.x + threadIdx.x; i < n; i += gridDim.x * blockDim.x)
        p[i] = 0u;
}

__global__ void init_labels_kernel(int* lab) {
    int i = blockIdx.x * blockDim.x + threadIdx.x;
    if (i < NEP) lab[i] = i;
}

// normalize line scores: ls = s / (1e-8 + max(s))
__global__ void ls_kernel(const float* __restrict__ lsIn, float* __restrict__ lsOut) {
    __shared__ float red[256];
    int t = threadIdx.x;
    float mx = -1e30f;
    for (int i = t; i < N_LINES; i += 256) mx = fmaxf(mx, lsIn[i]);
    red[t] = mx; __syncthreads();
    for (int off = 128; off > 0; off >>= 1) {
        if (t < off) red[t] = fmaxf(red[t], red[t + off]);
        __syncthreads();
    }
    float denom = 1e-8f + red[0];
    for (int i = t; i < N_LINES; i += 256) lsOut[i] = lsIn[i] / denom;
}

// ---------------- WMMA pairwise-distance kernels ----------------
// One wave (32 threads) per 16x16 tile. d^2 = |a|^2 + |b|^2 - 2 a.b,
// a.b computed by V_WMMA_F32_16X16X4_F32 with K padded 2->4 by zeros.
// A 16x4 f32 layout: lanes 0-15 M=0..15 hold K0/K1 in v[0]/v[1]; lanes 16-31 hold K2/K3 (=0).
// B 4x16 f32 layout: v[0]: lanes0-15 = row K0, lanes16-31 = row K2(=0); v[1]: K1 / K3(=0).
// C/D 16x16 f32: vgpr r: lanes0-15 = row r, lanes16-31 = row r+8; col = lane&15.

__global__ void adj_kernel(const float* __restrict__ ep, unsigned* __restrict__ adj) {
    const int ti = blockIdx.x * 16;     // row tile base (endpoint i)
    const int tj = blockIdx.y * 16;     // col tile base (endpoint j)
    const int lane = threadIdx.x;
    const int half = lane >> 4;
    const int l16  = lane & 15;
    const float keepA = (half == 0) ? 1.0f : 0.0f;

    int ra = ti + l16;
    int cb = tj + l16;
    v2f a, b;
    a[0] = ep[2 * ra + 0] * keepA; a[1] = ep[2 * ra + 1] * keepA;
    b[0] = ep[2 * cb + 0] * keepA; b[1] = ep[2 * cb + 1] * keepA;
    v8f c = {};
    c = __builtin_amdgcn_wmma_f32_16x16x4_f32(false, a, false, b, (short)0, c, false, false);

    float bx = ep[2 * cb + 0], by = ep[2 * cb + 1];
    float nB = bx * bx + by * by;               // norm of this lane's column
    unsigned m[8];
    int rowBase = ti + half * 8;
#pragma unroll
    for (int r = 0; r < 8; ++r) {
        int row = rowBase + r;
        float ax = ep[2 * row + 0], ay = ep[2 * row + 1];
        float d2 = ax * ax + ay * ay + nB - 2.0f * c[r];
        m[r] = __builtin_amdgcn_ballot_w32(d2 <= 9.0f);   // NMS_RADIUS^2
    }
    if (lane == 0) {
        int w  = tj >> 5;
        int sh = tj & 16;
#pragma unroll
        for (int r = 0; r < 8; ++r) {
            unsigned lo = m[r] & 0xFFFFu, hi = m[r] >> 16;
            if (lo) atomicOr(&adj[(ti + r) * ADJ_WORDS + w], lo << sh);
            if (hi) atomicOr(&adj[(ti + r + 8) * ADJ_WORDS + w], hi << sh);
        }
    }
}

__global__ void suppress_kernel(const float* __restrict__ kp, const float* __restrict__ ep,
                                unsigned* __restrict__ sup) {
    const int ti = blockIdx.x * 16;     // keypoint tile base
    const int tj = blockIdx.y * 16;     // endpoint tile base
    const int lane = threadIdx.x;
    const int half = lane >> 4;
    const int l16  = lane & 15;
    const float keepA = (half == 0) ? 1.0f : 0.0f;

    int ra = ti + l16;
    int cb = tj + l16;
    v2f a, b;
    a[0] = kp[2 * ra + 0] * keepA; a[1] = kp[2 * ra + 1] * keepA;
    b[0] = ep[2 * cb + 0] * keepA; b[1] = ep[2 * cb + 1] * keepA;
    v8f c = {};
    c = __builtin_amdgcn_wmma_f32_16x16x4_f32(false, a, false, b, (short)0, c, false, false);

    float bx = ep[2 * cb + 0], by = ep[2 * cb + 1];
    float nB = bx * bx + by * by;
    unsigned m[8];
    int rowBase = ti + half * 8;
#pragma unroll
    for (int r = 0; r < 8; ++r) {
        int row = rowBase + r;
        float ax = kp[2 * row + 0], ay = kp[2 * row + 1];
        float d2 = ax * ax + ay * ay + nB - 2.0f * c[r];
        m[r] = __builtin_amdgcn_ballot_w32(d2 < 16.0f);   // SP_NMS_RADIUS^2, strict
    }
    if (lane == 0) {
#pragma unroll
        for (int r = 0; r < 8; ++r) {
            if (m[r] & 0xFFFFu) atomicOr(&sup[ti + r], 1u);
            if (m[r] >> 16)     atomicOr(&sup[ti + r + 8], 1u);
        }
    }
}

// ---------------- connected components (Jacobi min-propagation) ----------------
__global__ void prop_kernel(const unsigned* __restrict__ adj,
                            const int* __restrict__ labIn, int* __restrict__ labOut) {
    __shared__ int red[32];
    int row  = blockIdx.x;
    int lane = threadIdx.x;
    int best = 0x7FFFFFFF;                 // self bit is always set (d2=0), so covered
    for (int w = lane; w < ADJ_WORDS; w += 32) {
        unsigned bits = adj[row * ADJ_WORDS + w];
        while (bits) {
            int b = __builtin_ctz(bits);
            bits &= bits - 1;
            int l = labIn[w * 32 + b];
            best = best < l ? best : l;
        }
    }
    red[lane] = best; __syncthreads();
    for (int off = 16; off > 0; off >>= 1) {
        if (lane < off) red[lane] = red[lane] < red[lane + off] ? red[lane] : red[lane + off];
        __syncthreads();
    }
    if (lane == 0) labOut[row] = red[0];
}

// dense rank of roots -> clusters. Single block of 1024 threads, 4 elems each.
__global__ void rank_kernel(const int* __restrict__ roots,
                            int* __restrict__ rankBuf, int* __restrict__ clusters) {
    __shared__ int s[1024];
    int t = threadIdx.x;
    int base = t * 4;
    int pre[4]; int sum = 0;
#pragma unroll
    for (int k = 0; k < 4; ++k) {
        int i = base + k;
        int p = (roots[i] == i) ? 1 : 0;
        pre[k] = sum; sum += p;
    }
    s[t] = sum; __syncthreads();
    for (int off = 1; off < 1024; off <<= 1) {
        int v = 0;
        if (t >= off) v = s[t - off];
        __syncthreads();
        s[t] += v;
        __syncthreads();
    }
    int offset = (t > 0) ? s[t - 1] : 0;   // exclusive prefix of this thread's chunk
#pragma unroll
    for (int k = 0; k < 4; ++k) rankBuf[base + k] = offset + pre[k];
    __threadfence_block();
    __syncthreads();
#pragma unroll
    for (int k = 0; k < 4; ++k) {
        int i = base + k;
        clusters[i] = rankBuf[roots[i]];
    }
}

// ---------------- deterministic segment means ----------------
__global__ void segsum_kernel(const float* __restrict__ ep, const float* __restrict__ ls,
                              const int* __restrict__ clusters,
                              float* __restrict__ junc, float* __restrict__ outPoints,
                              float* __restrict__ outScores) {
    int c = blockIdx.x * blockDim.x + threadIdx.x;   // cluster slot 0..4095
    if (c >= NEP) return;
    float sx = 0.f, sy = 0.f, ss = 0.f; int cnt = 0;
    for (int i = 0; i < NEP; ++i) {
        if (clusters[i] == c) {
            sx += ep[2 * i]; sy += ep[2 * i + 1];
            ss += ls[i >> 1];                        // ep_scores = repeat(ls, 2)
            ++cnt;
        }
    }
    float denom = cnt > 0 ? (float)cnt : 1.0f;
    float jx = sx / denom, jy = sy / denom, js = ss / denom;
    junc[2 * c] = jx; junc[2 * c + 1] = jy;
    outPoints[2 * c] = jx; outPoints[2 * c + 1] = jy;
    outScores[c] = js;
}

__global__ void newlines_kernel(const int* __restrict__ clusters, const float* __restrict__ junc,
                                float* __restrict__ outNewLines, float* __restrict__ outIdx) {
    int e = blockIdx.x * blockDim.x + threadIdx.x;   // endpoint 0..4095
    if (e >= NEP) return;
    int c = clusters[e];
    outNewLines[2 * e]     = junc[2 * c];
    outNewLines[2 * e + 1] = junc[2 * c + 1];
    outIdx[e] = (float)c;
}

// ---------------- junction descriptor sampling (bilinear + L2 norm) ----------------
__global__ void sample_kernel(const float* __restrict__ junc, const float* __restrict__ allDesc,
                              float* __restrict__ outDesc) {
    __shared__ float red[256];
    int m = blockIdx.x;          // junction slot
    int ch = threadIdx.x;        // channel
    float px = junc[2 * m], py = junc[2 * m + 1];
    const float s = 8.0f;
    const float scale = (float)WC * s - s * 0.5f - 0.5f;   // 1019.5
    float gx = ((px - s * 0.5f + 0.5f) / scale) * 2.0f - 1.0f;
    float gy = ((py - s * 0.5f + 0.5f) / scale) * 2.0f - 1.0f;
    float x = (gx + 1.0f) * 0.5f * (float)(WC - 1);
    float y = (gy + 1.0f) * 0.5f * (float)(HC - 1);
    float x0 = fminf(fmaxf(floorf(x), 0.0f), (float)(WC - 1));
    float y0 = fminf(fmaxf(floorf(y), 0.0f), (float)(HC - 1));
    float x1 = fminf(x0 + 1.0f, (float)(WC - 1));
    float y1 = fminf(y0 + 1.0f, (float)(HC - 1));
    float wx = x - x0, wy = y - y0;
    int x0i = (int)x0, x1i = (int)x1, y0i = (int)y0, y1i = (int)y1;
    const float* d = allDesc + ch * (HC * WC);
    float v = d[y0i * WC + x0i] * (1.f - wx) * (1.f - wy)
            + d[y0i * WC + x1i] * wx        * (1.f - wy)
            + d[y1i * WC + x0i] * (1.f - wx) * wy
            + d[y1i * WC + x1i] * wx        * wy;
    red[ch] = v * v; __syncthreads();
    for (int off = 128; off > 0; off >>= 1) {
        if (ch < off) red[ch] += red[ch + off];
        __syncthreads();
    }
    float nrm = fmaxf(sqrtf(red[0]), 1e-12f);
    outDesc[ch * NALL + m] = v / nrm;
}

// ---------------- keypoint passthrough (masked) ----------------
__global__ void kp_kernel(const float* __restrict__ kp, const float* __restrict__ ksc,
                          const float* __restrict__ desc, const unsigned* __restrict__ sup,
                          float* __restrict__ outPoints, float* __restrict__ outScores,
                          float* __restrict__ outDesc) {
    int k = blockIdx.x;          // keypoint
    int ch = threadIdx.x;        // channel
    bool keep = (sup[k] == 0u);
    int m = NEP + k;
    outDesc[ch * NALL + m] = keep ? desc[ch * NKP + k] : 0.0f;
    if (ch == 0) {
        outPoints[2 * m]     = keep ? kp[2 * k]     : 0.0f;
        outPoints[2 * m + 1] = keep ? kp[2 * k + 1] : 0.0f;
        outScores[m]         = keep ? ksc[k]        : 0.0f;
    }
}

// ---------------- host launcher ----------------
extern "C" void kernel_launch(void* const* d_in, const int* in_sizes, int n_in,
                              void* d_out, int out_size, void* d_ws, size_t ws_size,
                              hipStream_t stream) {
    const float* lines     = (const float*)d_in[0];   // (1,2048,2,2) == ep (4096,2)
    const float* lscores   = (const float*)d_in[1];   // (1,2048)
    const float* keypoints = (const float*)d_in[2];   // (1,2048,2)
    const float* kpscores  = (const float*)d_in[3];   // (1,2048)
    const float* desc      = (const float*)d_in[4];   // (1,256,2048)
    const float* allDesc   = (const float*)d_in[5];   // (1,256,128,128)

    float* out = (float*)d_out;
    char*  ws  = (char*)d_ws;

    unsigned* adj  = (unsigned*)(ws + WS_ADJ);
    unsigned* sup  = (unsigned*)(ws + WS_SUP);
    int*      lab0 = (int*)(ws + WS_LAB0);
    int*      lab1 = (int*)(ws + WS_LAB1);
    int*      rnk  = (int*)(ws + WS_RANK);
    int*      clus = (int*)(ws + WS_CLUS);
    float*    junc = (float*)(ws + WS_JUNC);

    float* outPoints   = out + OUT_POINTS;
    float* outScores   = out + OUT_SCORES;
    float* outDescs    = out + OUT_DESCS;
    float* outNewLines = out + OUT_NEWLINES;
    float* outIdx      = out + OUT_IDX;
    float* outLs       = out + OUT_LS;

    // 1) zero adjacency bitmask + suppression flags
    zero_kernel<<<1024, 256, 0, stream>>>(adj, NEP * ADJ_WORDS + NKP);

    // 2) normalized line scores
    ls_kernel<<<1, 256, 0, stream>>>(lscores, outLs);

    // 3) WMMA pairwise distances: keypoint suppression (2048 x 4096)
    suppress_kernel<<<dim3(NKP / 16, NEP / 16), 32, 0, stream>>>(keypoints, lines, sup);

    // 4) WMMA pairwise distances: DBSCAN adjacency (4096 x 4096)
    adj_kernel<<<dim3(NEP / 16, NEP / 16), 32, 0, stream>>>(lines, adj);

    // 5) connected components: init + 20 Jacobi min-propagation sweeps (ping-pong)
    init_labels_kernel<<<NEP / 256, 256, 0, stream>>>(lab0);
    for (int it = 0; it < 20; ++it) {
        const int* in  = (it & 1) ? lab1 : lab0;
        int*       o   = (it & 1) ? lab0 : lab1;
        prop_kernel<<<NEP, 32, 0, stream>>>(adj, in, o);
    }
    // 20 iterations (even) -> result in lab0

    // 6) dense relabel
    rank_kernel<<<1, 1024, 0, stream>>>(lab0, rnk, clus);

    // 7) deterministic segment means -> junctions / scores (also fills output slots)
    segsum_kernel<<<NEP / 256, 256, 0, stream>>>(lines, outLs, clus, junc, outPoints, outScores);

    // 8) new_lines + lines_junc_idx
    newlines_kernel<<<NEP / 256, 256, 0, stream>>>(clus, junc, outNewLines, outIdx);

    // 9) junction descriptor sampling -> all_descs[:, 0:4096]
    sample_kernel<<<NEP, NCH, 0, stream>>>(junc, allDesc, outDescs);

    // 10) masked keypoint passthrough -> all_* [4096:6144]
    kp_kernel<<<NKP, NCH, 0, stream>>>(keypoints, kpscores, desc, sup,
                                       outPoints, outScores, outDescs);
}